// SAGE_3350074490962
// MI455X (gfx1250) — compile-verified
//
#include <hip/hip_runtime.h>
#include <hip/hip_bf16.h>

typedef __attribute__((ext_vector_type(2))) float v2f;
typedef __attribute__((ext_vector_type(8))) float v8f;

#define H_DIM 256

// ---------------------------------------------------------------------------
// Zero a contiguous float region (agg1|cnt1|agg2|cnt2 in workspace).
// ---------------------------------------------------------------------------
__global__ void sage_zero_kernel(float* __restrict__ p, int n) {
  int i = blockIdx.x * blockDim.x + threadIdx.x;
  if (i < n) p[i] = 0.0f;
}

// ---------------------------------------------------------------------------
// One wave32 per edge: coalesced float4 row gather (x is 102MB < 192MB L2, so
// gathers hit L2), per-lane global_atomic_add_f32 scatter into agg, lane0
// bumps cnt.
// ---------------------------------------------------------------------------
__global__ __launch_bounds__(256) void sage_scatter_accum(
    const float* __restrict__ X, const int* __restrict__ src,
    const int* __restrict__ dst, float* __restrict__ agg,
    float* __restrict__ cnt, int E, int C) {
  unsigned gtid = blockIdx.x * blockDim.x + threadIdx.x;
  int e = (int)(gtid >> 5);
  int lane = (int)(gtid & 31u);
  if (e >= E) return;
  int s = src[e];
  int d = dst[e];
  const float4* xs = (const float4*)(X + (size_t)s * C);
  float* ad = agg + (size_t)d * C;
  int nv = C >> 2;  // float4s per row (32 for C=128, 64 for C=256)
  for (int c4 = lane; c4 < nv; c4 += 32) {
    float4 v = xs[c4];
    atomicAdd(ad + c4 * 4 + 0, v.x);
    atomicAdd(ad + c4 * 4 + 1, v.y);
    atomicAdd(ad + c4 * 4 + 2, v.z);
    atomicAdd(ad + c4 * 4 + 3, v.w);
  }
  if (lane == 0) atomicAdd(cnt + d, 1.0f);
}

// ---------------------------------------------------------------------------
// Fused SAGE layer GEMM: Out = act( (Aagg/max(cnt,1)) @ Wl + Xdst @ Wr + b ).
//
// Block = 256 threads (8 waves), covering a 64-row x 256-col output macro-tile
// (4 row-tiles x 16 col-tiles of 16x16). Each wave owns a 32-col strip
// (2 col-tiles). Per k-step a wave loads ONE v2f B fragment and feeds it to
// FOUR v_wmma_f32_16x16x4_f32 ops (one per row-tile) with independent
// accumulators: 4x fewer global B loads per WMMA than a 16-row block, and no
// D->C back-to-back chaining on consecutive WMMAs.
//
// A tiles (mean-scaled agg, root features) are staged in LDS with +1 row pad
// (K multiple of 64 banks otherwise => 16-way conflicts).
// ---------------------------------------------------------------------------
template <int K1, int K2, bool RELU>
__global__ __launch_bounds__(256) void sage_gemm(
    const float* __restrict__ Aagg,   // [M, K1]
    const float* __restrict__ cnt,    // [M]
    const float* __restrict__ Xdst,   // [M, K2] (first M rows of feature buf)
    const float* __restrict__ Wl,     // [K1, 256]
    const float* __restrict__ Wr,     // [K2, 256]
    const float* __restrict__ bias,   // [256]
    float* __restrict__ Out,          // [M, 256]
    int M) {
  constexpr int ROWS = 64;  // 4 row-tiles of 16
  constexpr int KP1 = K1 + 1;
  constexpr int KP2 = K2 + 1;
  __shared__ float sInv[ROWS];
  __shared__ float sA[ROWS * KP1];
  __shared__ float sX[ROWS * KP2];

  const int tid = threadIdx.x;
  const int row0 = blockIdx.x * ROWS;

  if (tid < ROWS) {
    int r = row0 + tid;
    float c = (r < M) ? cnt[r] : 1.0f;
    sInv[tid] = 1.0f / fmaxf(c, 1.0f);
  }
  __syncthreads();

  // Stage A (mean-scaled neighbor aggregate) and X (root features) into LDS.
  // Out-of-range rows are zero-filled (their stores are skipped anyway).
  for (int idx = tid; idx < ROWS * K1; idx += 256) {
    int r = idx / K1, k = idx - r * K1;
    float v = (row0 + r < M) ? Aagg[(size_t)(row0 + r) * K1 + k] * sInv[r]
                             : 0.0f;
    sA[r * KP1 + k] = v;
  }
  for (int idx = tid; idx < ROWS * K2; idx += 256) {
    int r = idx / K2, k = idx - r * K2;
    float v = (row0 + r < M) ? Xdst[(size_t)(row0 + r) * K2 + k] : 0.0f;
    sX[r * KP2 + k] = v;
  }
  __syncthreads();

  const int wave = tid >> 5;
  const int lane = tid & 31;
  const int mn = lane & 15;          // A row-in-tile / B col-in-tile
  const int koff = (lane >> 4) * 2;  // k sub-offset within each group of 4

  for (int t = 0; t < 2; ++t) {
    const int col = (wave * 2 + t) * 16 + mn;
    v8f acc[4] = {v8f{}, v8f{}, v8f{}, v8f{}};

#pragma unroll 4
    for (int kk = 0; kk < K1; kk += 4) {
      v2f b;
      b.x = Wl[(size_t)(kk + koff) * H_DIM + col];
      b.y = Wl[(size_t)(kk + koff + 1) * H_DIM + col];
#pragma unroll
      for (int mt = 0; mt < 4; ++mt) {
        v2f a;
        a.x = sA[(mt * 16 + mn) * KP1 + kk + koff];
        a.y = sA[(mt * 16 + mn) * KP1 + kk + koff + 1];
        acc[mt] = __builtin_amdgcn_wmma_f32_16x16x4_f32(
            false, a, false, b, (short)0, acc[mt], false, false);
      }
    }
#pragma unroll 4
    for (int kk = 0; kk < K2; kk += 4) {
      v2f b;
      b.x = Wr[(size_t)(kk + koff) * H_DIM + col];
      b.y = Wr[(size_t)(kk + koff + 1) * H_DIM + col];
#pragma unroll
      for (int mt = 0; mt < 4; ++mt) {
        v2f a;
        a.x = sX[(mt * 16 + mn) * KP2 + kk + koff];
        a.y = sX[(mt * 16 + mn) * KP2 + kk + koff + 1];
        acc[mt] = __builtin_amdgcn_wmma_f32_16x16x4_f32(
            false, a, false, b, (short)0, acc[mt], false, false);
      }
    }

    // Epilogue: bias (+ReLU). D layout: VGPR v -> row = v + 8*(lane>=16).
    const float bv = bias[col];
    const int mbase = (lane >> 4) * 8;
#pragma unroll
    for (int mt = 0; mt < 4; ++mt) {
#pragma unroll
      for (int v = 0; v < 8; ++v) {
        int row = row0 + mt * 16 + mbase + v;
        if (row < M) {
          float val = acc[mt][v] + bv;
          if (RELU) val = fmaxf(val, 0.0f);
          Out[(size_t)row * H_DIM + col] = val;
        }
      }
    }
  }
}

// ---------------------------------------------------------------------------
// Launch: zero -> scatter1 -> gemm1(relu) -> scatter2 -> gemm2 -> d_out.
// Stream ordering provides all cross-kernel dependencies.
// ---------------------------------------------------------------------------
extern "C" void kernel_launch(void* const* d_in, const int* in_sizes, int n_in,
                              void* d_out, int out_size, void* d_ws,
                              size_t ws_size, hipStream_t stream) {
  const float* x   = (const float*)d_in[0];   // [N0, 128]
  const int*  src1 = (const int*)d_in[1];     // [E1]
  const int*  dst1 = (const int*)d_in[2];     // [E1]
  const int*  src2 = (const int*)d_in[3];     // [E2]
  const int*  dst2 = (const int*)d_in[4];     // [E2]
  const float* Wl1 = (const float*)d_in[5];   // [128, 256]
  const float* Wr1 = (const float*)d_in[6];   // [128, 256]
  const float* b1  = (const float*)d_in[7];   // [256]
  const float* Wl2 = (const float*)d_in[8];   // [256, 256]
  const float* Wr2 = (const float*)d_in[9];   // [256, 256]
  const float* b2  = (const float*)d_in[10];  // [256]
  float* out = (float*)d_out;                 // [N2, 256]

  const int N1 = 50000, N2 = 10000;
  const int E1 = 1000000, E2 = 250000;

  float* ws   = (float*)d_ws;
  float* agg1 = ws;                                // N1*128
  float* cnt1 = agg1 + (size_t)N1 * 128;           // N1
  float* agg2 = cnt1 + N1;                         // N2*256
  float* cnt2 = agg2 + (size_t)N2 * 256;           // N2
  float* h    = cnt2 + N2;                         // N1*256

  // Zero the accumulator + count regions (contiguous prefix of ws).
  const int nzero = N1 * 128 + N1 + N2 * 256 + N2;
  sage_zero_kernel<<<(nzero + 255) / 256, 256, 0, stream>>>(ws, nzero);

  // Layer 1 aggregation: one wave per edge.
  {
    unsigned nthreads = (unsigned)E1 * 32u;
    sage_scatter_accum<<<(nthreads + 255u) / 256u, 256, 0, stream>>>(
        x, src1, dst1, agg1, cnt1, E1, 128);
  }

  // Layer 1 fused GEMM + bias + ReLU -> h.
  sage_gemm<128, 128, true><<<(N1 + 63) / 64, 256, 0, stream>>>(
      agg1, cnt1, x, Wl1, Wr1, b1, h, N1);

  // Layer 2 aggregation over h.
  {
    unsigned nthreads = (unsigned)E2 * 32u;
    sage_scatter_accum<<<(nthreads + 255u) / 256u, 256, 0, stream>>>(
        h, src2, dst2, agg2, cnt2, E2, 256);
  }

  // Layer 2 fused GEMM + bias -> output.
  sage_gemm<256, 256, false><<<(N2 + 63) / 64, 256, 0, stream>>>(
      agg2, cnt2, h, Wl2, Wr2, b2, out, N2);
}